// ChebyshevGraphConv_13451837571469
// MI455X (gfx1250) — compile-verified
//
#include <hip/hip_runtime.h>
#include <float.h>

// Problem sizes (fixed by the reference)
constexpr int kB = 4;
constexpr int kN = 256;
constexpr int kC = 128;
constexpr float kEPS = 1e-5f;
constexpr float kLNEPS = 1e-5f;

typedef float v2f __attribute__((ext_vector_type(2)));
typedef float v8f __attribute__((ext_vector_type(8)));

// ---------------------------------------------------------------------------
// helpers
// ---------------------------------------------------------------------------
__device__ __forceinline__ void atomicMaxF(float* addr, float v) {
  // order-preserving int/uint trick; valid for non-NaN floats
  if (v >= 0.0f) {
    atomicMax((int*)addr, __float_as_int(v));
  } else {
    atomicMin((unsigned int*)addr, __float_as_uint(v));
  }
}

__device__ __forceinline__ v2f ld2(const float* p) { return *(const v2f*)p; }

// Two 16x16 output tiles (rows A0 / A1), K=128, f32 WMMA.
// B fragments are shared across both accumulators; all fragment loads are
// software-pipelined one k-step ahead so DS latency overlaps the WMMAs.
// lid = lane&15, koff = 2*(lane>>4) per the 16x4 f32 A-operand layout.
__device__ __forceinline__ void wmma_k128_x2(const float* __restrict__ A0,
                                             const float* __restrict__ A1, int lda,
                                             const float* __restrict__ Bw, int ldb,
                                             int lid, int koff, v8f& acc0, v8f& acc1) {
  v2f a0 = ld2(A0 + lid * lda + koff);
  v2f a1 = ld2(A1 + lid * lda + koff);
  v2f b;
  b.x = Bw[koff * ldb + lid];
  b.y = Bw[(koff + 1) * ldb + lid];
#pragma unroll
  for (int k0 = 0; k0 < kC; k0 += 4) {
    v2f ca0 = a0, ca1 = a1, cb = b;
    int kn = k0 + 4;
    if (kn < kC) {
      a0 = ld2(A0 + lid * lda + kn + koff);
      a1 = ld2(A1 + lid * lda + kn + koff);
      b.x = Bw[(kn + koff) * ldb + lid];
      b.y = Bw[(kn + koff + 1) * ldb + lid];
    }
    acc0 = __builtin_amdgcn_wmma_f32_16x16x4_f32(false, ca0, false, cb, (short)0, acc0,
                                                 false, false);
    acc1 = __builtin_amdgcn_wmma_f32_16x16x4_f32(false, ca1, false, cb, (short)0, acc1,
                                                 false, false);
  }
}

// ---------------------------------------------------------------------------
// 1) degree -> dinv, then Laplacian
// ---------------------------------------------------------------------------
__global__ void k_dinv(const float* __restrict__ adj, float* __restrict__ dinv) {
  int b = blockIdx.x;
  int j = threadIdx.x;  // 256 threads = one column each
  const float* A = adj + b * kN * kN;
  float s = 0.0f;
  for (int i = 0; i < kN; ++i) s += A[i * kN + j];
  dinv[b * kN + j] = (s == 0.0f) ? 1.0f : rsqrtf(s);
}

__global__ void k_lap(const float* __restrict__ adj, const float* __restrict__ dinv,
                      float* __restrict__ lap) {
  int b = blockIdx.y;
  int idx = blockIdx.x * blockDim.x + threadIdx.x;  // over N*N
  int i = idx >> 8, j = idx & 255;
  float v = dinv[b * kN + i] * adj[b * kN * kN + idx] * dinv[b * kN + j];
  lap[b * kN * kN + idx] = (i == j ? 1.0f : 0.0f) - v;
}

// ---------------------------------------------------------------------------
// 2) spectral norm via power iteration on L^T L (L resident in LDS), then
//    cheb_lap = 2*L/(sigma+eps) - I  written straight from LDS.
//    LDS: 256 x 257 (padded) + 3*256 scratch = 260 KB < 320 KB/WGP.
// ---------------------------------------------------------------------------
__global__ void k_power_cheb(const float* __restrict__ lap, float* __restrict__ cl,
                             float* __restrict__ sigma_out) {
  extern __shared__ float sm[];
  float* sL = sm;                 // [256][257]
  float* sV = sm + kN * 257;      // [256]
  float* sW = sV + kN;            // [256]
  float* sRed = sW + kN;          // [256]
  const int b = blockIdx.x;
  const int t = threadIdx.x;
  const float* L = lap + b * kN * kN;
  for (int idx = t; idx < kN * kN; idx += blockDim.x)
    sL[(idx >> 8) * 257 + (idx & 255)] = L[idx];
  sV[t] = 1.0f;
  __syncthreads();

  for (int it = 0; it < 96; ++it) {
    // w = L v
    float s = 0.0f;
    const float* row = sL + t * 257;
    for (int j = 0; j < kN; ++j) s += row[j] * sV[j];
    sW[t] = s;
    __syncthreads();
    // u = L^T w (column walk is contiguous across threads)
    float s2 = 0.0f;
    for (int i = 0; i < kN; ++i) s2 += sL[i * 257 + t] * sW[i];
    sRed[t] = s2 * s2;
    __syncthreads();
    for (int off = 128; off > 0; off >>= 1) {
      if (t < off) sRed[t] += sRed[t + off];
      __syncthreads();
    }
    float inv = rsqrtf(fmaxf(sRed[0], 1e-30f));
    __syncthreads();
    sV[t] = s2 * inv;
    __syncthreads();
  }
  // sigma^2 = ||L v||^2 with ||v|| = 1
  {
    float s = 0.0f;
    const float* row = sL + t * 257;
    for (int j = 0; j < kN; ++j) s += row[j] * sV[j];
    sRed[t] = s * s;
  }
  __syncthreads();
  for (int off = 128; off > 0; off >>= 1) {
    if (t < off) sRed[t] += sRed[t + off];
    __syncthreads();
  }
  float sigma = sqrtf(fmaxf(sRed[0], 0.0f));
  if (t == 0) sigma_out[b] = sigma;
  float scale = 2.0f / (sigma + kEPS);
  float* CL = cl + b * kN * kN;
  for (int idx = t; idx < kN * kN; idx += blockDim.x) {
    int i = idx >> 8, j = idx & 255;
    CL[idx] = scale * sL[i * 257 + j] - (i == j ? 1.0f : 0.0f);
  }
}

// ---------------------------------------------------------------------------
// 3) small precomputes: mg = graph@W_mg + b_mg + b_me ; msg1/msg2c/t1 ; h
// ---------------------------------------------------------------------------
__global__ void k_mg(const float* __restrict__ graph, const float* __restrict__ W_mg,
                     const float* __restrict__ b_mg, const float* __restrict__ b_me,
                     float* __restrict__ mg) {
  int b = blockIdx.x, c = threadIdx.x;
  float s = b_mg[c] + b_me[c];
  for (int k = 0; k < kC; ++k) s += graph[b * kC + k] * W_mg[k * kC + c];
  mg[b * kC + c] = s;
}

__global__ void k_msg12(const float* __restrict__ node, const float* __restrict__ hid,
                        const float* __restrict__ W_m1, const float* __restrict__ b_m1,
                        const float* __restrict__ W_m2, const float* __restrict__ b_m2,
                        const float* __restrict__ W_fe1, const float* __restrict__ b_fe1,
                        const float* __restrict__ mg, float* __restrict__ msg1,
                        float* __restrict__ msg2c, float* __restrict__ t1) {
  __shared__ float sZ[2 * kC];
  int r = blockIdx.x;  // b*N + n
  int b = r >> 8;
  int c = threadIdx.x;  // 128
  sZ[c] = node[r * kC + c];
  sZ[kC + c] = hid[r * kC + c];
  __syncthreads();
  float s1 = b_m1[c], s2 = b_m2[c], s3 = b_fe1[c];
  for (int k = 0; k < 2 * kC; ++k) {
    float zv = sZ[k];
    s1 += zv * W_m1[k * kC + c];
    s2 += zv * W_m2[k * kC + c];
    s3 += zv * W_fe1[k * kC + c];
  }
  msg1[r * kC + c] = s1;
  msg2c[r * kC + c] = s2 + mg[b * kC + c];  // + graph msg + b_me folded in
  t1[r * kC + c] = fmaxf(s3, 0.0f);
}

__global__ void k_h(const float* __restrict__ t1, const float* __restrict__ W_fe2,
                    const float* __restrict__ b_fe2, float* __restrict__ h) {
  __shared__ float sT[kC];
  int r = blockIdx.x;
  int c = threadIdx.x;
  sT[c] = t1[r * kC + c];
  __syncthreads();
  float s = b_fe2[c];
  for (int k = 0; k < kC; ++k) s += sT[k] * W_fe2[k * kC + c];
  h[r * kC + c] = s;
}

__global__ void k_fill(float* __restrict__ p, float v, int n) {
  int i = blockIdx.x * blockDim.x + threadIdx.x;
  if (i < n) p[i] = v;
}

// ---------------------------------------------------------------------------
// 4) BIG fused message kernel (WMMA): per block = (b, 16-j tile, 64-i chunk).
//    chain  E@W_me (+msg1[j]+msg2c[i]) -> relu -> @W_mlp1 -> relu -> @W_mlp2
//    running max over i in registers, atomic-max into msgs.
//    LDS: 3 weights [128][132] + E/T staging [64][132]x2 + msg tiles = 274 KB.
// ---------------------------------------------------------------------------
__global__ __launch_bounds__(256) void k_msgs_big(
    const float* __restrict__ edge, const float* __restrict__ msg1g,
    const float* __restrict__ msg2g, const float* __restrict__ b_mlp1,
    const float* __restrict__ b_mlp2, const float* __restrict__ W_me,
    const float* __restrict__ W_mlp1, const float* __restrict__ W_mlp2,
    float* __restrict__ msgs) {
  extern __shared__ float sm[];
  float* sWme = sm;                  // [128][132]
  float* sW1 = sWme + kC * 132;      // [128][132]
  float* sW2 = sW1 + kC * 132;       // [128][132]
  float* sE = sW2 + kC * 132;        // [64][132]
  float* sT = sE + 64 * 132;         // [64][132]
  float* sM1 = sT + 64 * 132;        // [16][128]
  float* sM2 = sM1 + 16 * kC;        // [4][128]

  const int t = threadIdx.x;
  const int jt = blockIdx.x;  // 16 j-tiles
  const int ic = blockIdx.y;  // 4 i-chunks of 64
  const int b = blockIdx.z;
  const int j0 = jt * 16;
  const int lane = t & 31;
  const int lid = lane & 15;
  const int half = lane >> 4;
  const int koff = half * 2;
  const int wv = t >> 5;      // 8 waves
  const int n0 = wv * 16;

  // stage the three weight matrices as float4 (16 x b128 per thread)
  for (int idx = t; idx < (kC * kC) / 4; idx += 256) {
    int k = idx >> 5, n4 = (idx & 31) * 4;
    *(float4*)&sWme[k * 132 + n4] = ((const float4*)W_me)[idx];
    *(float4*)&sW1[k * 132 + n4] = ((const float4*)W_mlp1)[idx];
    *(float4*)&sW2[k * 132 + n4] = ((const float4*)W_mlp2)[idx];
  }
  for (int idx = t; idx < (16 * kC) / 4; idx += 256) {
    int jl = idx >> 5, c4 = (idx & 31) * 4;
    *(float4*)&sM1[jl * kC + c4] =
        ((const float4*)(msg1g + ((b * kN) + j0 + jl) * kC))[idx & 31];
  }
  const float bm1v = b_mlp1[n0 + lid];
  const float bm2v = b_mlp2[n0 + lid];

  v8f vmax;
#pragma unroll
  for (int v = 0; v < 8; ++v) vmax[v] = -FLT_MAX;

  for (int s = 0; s < 16; ++s) {
    const int i0 = ic * 64 + s * 4;
    // stage E tile as float4: rows r = il*16 + jl
    for (int idx = t; idx < (64 * kC) / 4; idx += 256) {
      int r = idx >> 5, c4 = (idx & 31) * 4;
      int il = r >> 4, jl = r & 15;
      *(float4*)&sE[r * 132 + c4] =
          ((const float4*)(edge + (((b * kN) + (i0 + il)) * kN + (j0 + jl)) * kC))[idx & 31];
    }
    for (int idx = t; idx < (4 * kC) / 4; idx += 256) {
      int il = idx >> 5, c4 = (idx & 31) * 4;
      *(float4*)&sM2[il * kC + c4] =
          ((const float4*)(msg2g + ((b * kN) + i0 + il) * kC))[idx & 31];
    }
    __syncthreads();

    // GEMM1: T = relu(E@W_me + msg1[j] + msg2c[i])
#pragma unroll 1
    for (int m = 0; m < 4; m += 2) {
      v8f acc0 = {}, acc1 = {};
      wmma_k128_x2(sE + m * 16 * 132, sE + (m + 1) * 16 * 132, 132, sWme + n0, 132,
                   lid, koff, acc0, acc1);
#pragma unroll
      for (int v = 0; v < 8; ++v) {
        int jl = v + 8 * half;
        float v0 = acc0[v] + sM1[jl * kC + n0 + lid] + sM2[m * kC + n0 + lid];
        float v1 = acc1[v] + sM1[jl * kC + n0 + lid] + sM2[(m + 1) * kC + n0 + lid];
        sT[(m * 16 + jl) * 132 + n0 + lid] = fmaxf(v0, 0.0f);
        sT[((m + 1) * 16 + jl) * 132 + n0 + lid] = fmaxf(v1, 0.0f);
      }
    }
    __syncthreads();

    // GEMM2: V = relu(T@W_mlp1 + b_mlp1) -> reuse sE as V buffer
#pragma unroll 1
    for (int m = 0; m < 4; m += 2) {
      v8f acc0 = {}, acc1 = {};
      wmma_k128_x2(sT + m * 16 * 132, sT + (m + 1) * 16 * 132, 132, sW1 + n0, 132,
                   lid, koff, acc0, acc1);
#pragma unroll
      for (int v = 0; v < 8; ++v) {
        int jl = v + 8 * half;
        sE[(m * 16 + jl) * 132 + n0 + lid] = fmaxf(acc0[v] + bm1v, 0.0f);
        sE[((m + 1) * 16 + jl) * 132 + n0 + lid] = fmaxf(acc1[v] + bm1v, 0.0f);
      }
    }
    __syncthreads();

    // GEMM3: W = V@W_mlp2 + b_mlp2 -> running max over i
#pragma unroll 1
    for (int m = 0; m < 4; m += 2) {
      v8f acc0 = {}, acc1 = {};
      wmma_k128_x2(sE + m * 16 * 132, sE + (m + 1) * 16 * 132, 132, sW2 + n0, 132,
                   lid, koff, acc0, acc1);
#pragma unroll
      for (int v = 0; v < 8; ++v) {
        vmax[v] = fmaxf(vmax[v], fmaxf(acc0[v], acc1[v]) + bm2v);
      }
    }
    __syncthreads();
  }
#pragma unroll
  for (int v = 0; v < 8; ++v) {
    int jl = v + 8 * half;
    atomicMaxF(&msgs[((b * kN) + j0 + jl) * kC + n0 + lid], vmax[v]);
  }
}

// ---------------------------------------------------------------------------
// 5) edge_out (WMMA): out[b,i,j,:] = e@Ew0 + cl*(e@Ew1) + (2cl^2-1)*(e@Ew2)
//    2 M-tiles x 3 B-matrices per pass: each A/B fragment feeds multiple
//    WMMAs; loads pipelined one k-step ahead. LDS: 3 weights + E = 265 KB.
// ---------------------------------------------------------------------------
__global__ __launch_bounds__(256) void k_edge_out(const float* __restrict__ edge,
                                                  const float* __restrict__ cl,
                                                  const float* __restrict__ ew,
                                                  float* __restrict__ eo) {
  extern __shared__ float sm[];
  float* sW0 = sm;
  float* sW1 = sW0 + kC * 132;
  float* sW2 = sW1 + kC * 132;
  float* sE = sW2 + kC * 132;  // [128][132]
  float* sCL = sE + kC * 132;  // [128]
  const int t = threadIdx.x;
  const int jh = blockIdx.x;  // 0..1 (half of j)
  const int ig = blockIdx.y;  // 0..31 (8 i's each)
  const int b = blockIdx.z;
  const int j0 = jh * 128;
  const int lane = t & 31, lid = lane & 15, half = lane >> 4, koff = half * 2;
  const int wv = t >> 5, n0 = wv * 16;

  for (int idx = t; idx < (kC * kC) / 4; idx += 256) {
    int k = idx >> 5, n4 = (idx & 31) * 4;
    *(float4*)&sW0[k * 132 + n4] = ((const float4*)ew)[idx];
    *(float4*)&sW1[k * 132 + n4] = ((const float4*)(ew + kC * kC))[idx];
    *(float4*)&sW2[k * 132 + n4] = ((const float4*)(ew + 2 * kC * kC))[idx];
  }
  for (int il = 0; il < 8; ++il) {
    const int i = ig * 8 + il;
    for (int idx = t; idx < (128 * kC) / 4; idx += 256) {
      int r = idx >> 5, c4 = (idx & 31) * 4;
      *(float4*)&sE[r * 132 + c4] =
          ((const float4*)(edge + (((b * kN) + i) * kN + (j0 + r)) * kC))[idx & 31];
    }
    if (t < 32) {
      *(float4*)&sCL[t * 4] = ((const float4*)(cl + ((b * kN) + i) * kN + j0))[t];
    }
    __syncthreads();
#pragma unroll 1
    for (int m = 0; m < 8; m += 2) {
      v8f p00 = {}, p01 = {}, p02 = {};  // tile m     x Ew0/1/2
      v8f p10 = {}, p11 = {}, p12 = {};  // tile m+1   x Ew0/1/2
      const float* A0 = sE + (m * 16 + lid) * 132 + koff;
      const float* A1 = A0 + 16 * 132;
      v2f a0 = ld2(A0), a1 = ld2(A1);
      int bi = koff * 132 + n0 + lid;
      v2f b0, b1, b2;
      b0.x = sW0[bi]; b0.y = sW0[bi + 132];
      b1.x = sW1[bi]; b1.y = sW1[bi + 132];
      b2.x = sW2[bi]; b2.y = sW2[bi + 132];
#pragma unroll
      for (int k0 = 0; k0 < kC; k0 += 4) {
        v2f ca0 = a0, ca1 = a1, cb0 = b0, cb1 = b1, cb2 = b2;
        int kn = k0 + 4;
        if (kn < kC) {
          a0 = ld2(A0 + kn);
          a1 = ld2(A1 + kn);
          int bn = (kn + koff) * 132 + n0 + lid;
          b0.x = sW0[bn]; b0.y = sW0[bn + 132];
          b1.x = sW1[bn]; b1.y = sW1[bn + 132];
          b2.x = sW2[bn]; b2.y = sW2[bn + 132];
        }
        p00 = __builtin_amdgcn_wmma_f32_16x16x4_f32(false, ca0, false, cb0, (short)0, p00, false, false);
        p01 = __builtin_amdgcn_wmma_f32_16x16x4_f32(false, ca0, false, cb1, (short)0, p01, false, false);
        p02 = __builtin_amdgcn_wmma_f32_16x16x4_f32(false, ca0, false, cb2, (short)0, p02, false, false);
        p10 = __builtin_amdgcn_wmma_f32_16x16x4_f32(false, ca1, false, cb0, (short)0, p10, false, false);
        p11 = __builtin_amdgcn_wmma_f32_16x16x4_f32(false, ca1, false, cb1, (short)0, p11, false, false);
        p12 = __builtin_amdgcn_wmma_f32_16x16x4_f32(false, ca1, false, cb2, (short)0, p12, false, false);
      }
#pragma unroll
      for (int v = 0; v < 8; ++v) {
        int row0 = m * 16 + v + 8 * half;
        int row1 = row0 + 16;
        float s1a = sCL[row0];
        float s2a = 2.0f * s1a * s1a - 1.0f;
        float s1b = sCL[row1];
        float s2b = 2.0f * s1b * s1b - 1.0f;
        eo[(((b * kN) + i) * kN + (j0 + row0)) * kC + n0 + lid] =
            p00[v] + s1a * p01[v] + s2a * p02[v];
        eo[(((b * kN) + i) * kN + (j0 + row1)) * kC + n0 + lid] =
            p10[v] + s1b * p11[v] + s2b * p12[v];
      }
    }
    __syncthreads();
  }
}

// ---------------------------------------------------------------------------
// 6) Chebyshev matvecs: out = alpha * (cl @ X) - (Yprev ? Yprev : 0)
// ---------------------------------------------------------------------------
__global__ void k_chebmm(const float* __restrict__ cl, const float* __restrict__ X,
                         const float* __restrict__ Yprev, float alpha,
                         float* __restrict__ out) {
  __shared__ float sRow[kN];
  int r = blockIdx.x;  // b*N + i
  int b = r >> 8;
  int c = threadIdx.x;  // 128
  const float* CL = cl + r * kN;
  sRow[c] = CL[c];
  sRow[c + 128] = CL[c + 128];
  __syncthreads();
  const float* Xb = X + b * kN * kC;
  float s = 0.0f;
  for (int j = 0; j < kN; ++j) s += sRow[j] * Xb[j * kC + c];
  float o = alpha * s;
  if (Yprev) o -= Yprev[r * kC + c];
  out[r * kC + c] = o;
}

// ---------------------------------------------------------------------------
// 7) final fused: node_out, three output linears, layernorm
// ---------------------------------------------------------------------------
__global__ void k_final(const float* __restrict__ msgs, const float* __restrict__ h,
                        const float* __restrict__ cn1, const float* __restrict__ cn2,
                        const float* __restrict__ node, const float* __restrict__ hid,
                        const float* __restrict__ NW, const float* __restrict__ W_o1,
                        const float* __restrict__ b_o1, const float* __restrict__ W_o2,
                        const float* __restrict__ b_o2, const float* __restrict__ W_o3,
                        const float* __restrict__ b_o3, const float* __restrict__ ln_g,
                        const float* __restrict__ ln_b, float* __restrict__ out) {
  __shared__ float sMsgs[kC], sNode[kC], sZ[2 * kC], sRed[kC], sRed2[kC];
  int r = blockIdx.x;
  int c = threadIdx.x;  // 128
  sMsgs[c] = msgs[r * kC + c];
  sZ[c] = node[r * kC + c];
  sZ[kC + c] = hid[r * kC + c];
  __syncthreads();
  float p0 = 0.f, p1 = 0.f, p2 = 0.f;
  for (int k = 0; k < kC; ++k) {
    float m = sMsgs[k];
    p0 += m * NW[k * kC + c];
    p1 += m * NW[kC * kC + k * kC + c];
    p2 += m * NW[2 * kC * kC + k * kC + c];
  }
  sNode[c] = h[r * kC + c] * p0 + cn1[r * kC + c] * p1 + cn2[r * kC + c] * p2;
  __syncthreads();
  float o = b_o1[c] + b_o2[c] + b_o3[c];
  for (int k = 0; k < kC; ++k) o += sMsgs[k] * W_o1[k * kC + c] + sNode[k] * W_o2[k * kC + c];
  for (int k = 0; k < 2 * kC; ++k) o += sZ[k] * W_o3[k * kC + c];
  sRed[c] = o;
  sRed2[c] = o * o;
  __syncthreads();
  for (int off = 64; off > 0; off >>= 1) {
    if (c < off) { sRed[c] += sRed[c + off]; sRed2[c] += sRed2[c + off]; }
    __syncthreads();
  }
  float mu = sRed[0] * (1.0f / kC);
  float var = sRed2[0] * (1.0f / kC) - mu * mu;
  out[r * kC + c] = (o - mu) * rsqrtf(var + kLNEPS) * ln_g[c] + ln_b[c];
}

// ---------------------------------------------------------------------------
// launcher
// ---------------------------------------------------------------------------
extern "C" void kernel_launch(void* const* d_in, const int* in_sizes, int n_in,
                              void* d_out, int out_size, void* d_ws, size_t ws_size,
                              hipStream_t stream) {
  const float* node = (const float*)d_in[0];
  const float* edge = (const float*)d_in[1];
  const float* graph = (const float*)d_in[2];
  const float* hid = (const float*)d_in[3];
  const float* adj = (const float*)d_in[4];
  const float* W_m1 = (const float*)d_in[5];
  const float* b_m1 = (const float*)d_in[6];
  const float* W_m2 = (const float*)d_in[7];
  const float* b_m2 = (const float*)d_in[8];
  const float* W_me = (const float*)d_in[9];
  const float* b_me = (const float*)d_in[10];
  const float* W_mg = (const float*)d_in[11];
  const float* b_mg = (const float*)d_in[12];
  const float* W_fe1 = (const float*)d_in[13];
  const float* b_fe1 = (const float*)d_in[14];
  const float* W_fe2 = (const float*)d_in[15];
  const float* b_fe2 = (const float*)d_in[16];
  const float* W_mlp1 = (const float*)d_in[17];
  const float* b_mlp1 = (const float*)d_in[18];
  const float* W_mlp2 = (const float*)d_in[19];
  const float* b_mlp2 = (const float*)d_in[20];
  const float* node_w = (const float*)d_in[21];
  const float* edge_w = (const float*)d_in[22];
  const float* W_o1 = (const float*)d_in[23];
  const float* b_o1 = (const float*)d_in[24];
  const float* W_o2 = (const float*)d_in[25];
  const float* b_o2 = (const float*)d_in[26];
  const float* W_o3 = (const float*)d_in[27];
  const float* b_o3 = (const float*)d_in[28];
  const float* ln_g = (const float*)d_in[29];
  const float* ln_b = (const float*)d_in[30];

  float* out = (float*)d_out;                  // [B,N,C]
  float* edge_out = out + kB * kN * kC;        // [B,N,N,C]

  const int BNN = kB * kN * kN;   // 262144
  const int BNC = kB * kN * kC;   // 131072
  float* ws = (float*)d_ws;
  float* w_dinv = ws;             // 1024
  float* w_sigma = ws + 1024;     // 4 (padded to 2048)
  float* w_lap = ws + 2048;       // BNN
  float* w_cl = w_lap + BNN;      // BNN
  float* w_mg = w_cl + BNN;       // 512
  float* w_msg1 = w_mg + 512;     // BNC
  float* w_msg2c = w_msg1 + BNC;  // BNC
  float* w_t1 = w_msg2c + BNC;    // BNC
  float* w_h = w_t1 + BNC;        // BNC
  float* w_msgs = w_h + BNC;      // BNC
  float* w_cn1 = w_msgs + BNC;    // BNC
  float* w_cn2 = w_cn1 + BNC;     // BNC

  // 1) Laplacian
  k_dinv<<<kB, kN, 0, stream>>>(adj, w_dinv);
  k_lap<<<dim3(kN, kB), 256, 0, stream>>>(adj, w_dinv, w_lap);

  // 2) sigma + cheb_lap (L in LDS: 256*257 + 3*256 floats)
  size_t lds_pow = (size_t)(kN * 257 + 3 * kN) * sizeof(float);
  k_power_cheb<<<kB, 256, lds_pow, stream>>>(w_lap, w_cl, w_sigma);

  // 3) small precomputes
  k_mg<<<kB, kC, 0, stream>>>(graph, W_mg, b_mg, b_me, w_mg);
  k_msg12<<<kB * kN, kC, 0, stream>>>(node, hid, W_m1, b_m1, W_m2, b_m2, W_fe1, b_fe1,
                                      w_mg, w_msg1, w_msg2c, w_t1);
  k_h<<<kB * kN, kC, 0, stream>>>(w_t1, W_fe2, b_fe2, w_h);
  k_fill<<<BNC / 256, 256, 0, stream>>>(w_msgs, -FLT_MAX, BNC);

  // 4) big fused message kernel (WMMA)
  size_t lds_big = (size_t)(3 * kC * 132 + 2 * 64 * 132 + 16 * kC + 4 * kC) * sizeof(float);
  k_msgs_big<<<dim3(16, 4, kB), 256, lds_big, stream>>>(
      edge, w_msg1, w_msg2c, b_mlp1, b_mlp2, W_me, W_mlp1, W_mlp2, w_msgs);

  // 5) edge_out (WMMA)
  size_t lds_eo = (size_t)(4 * kC * 132 + 128) * sizeof(float);
  k_edge_out<<<dim3(2, 32, kB), 256, lds_eo, stream>>>(edge, w_cl, edge_w, edge_out);

  // 6) Chebyshev node features
  k_chebmm<<<kB * kN, kC, 0, stream>>>(w_cl, w_h, nullptr, 1.0f, w_cn1);
  k_chebmm<<<kB * kN, kC, 0, stream>>>(w_cl, w_cn1, w_h, 2.0f, w_cn2);

  // 7) final fused output + layernorm
  k_final<<<kB * kN, kC, 0, stream>>>(w_msgs, w_h, w_cn1, w_cn2, node, hid, node_w,
                                      W_o1, b_o1, W_o2, b_o2, W_o3, b_o3, ln_g, ln_b,
                                      out);
}